// SphereDownGeo_49392123904075
// MI455X (gfx1250) — compile-verified
//
#include <hip/hip_runtime.h>

typedef __attribute__((ext_vector_type(2))) float v2f;
typedef __attribute__((ext_vector_type(8))) float v8f;

#define NSIDE  512
#define N_IN   (12 * NSIDE * NSIDE)   // 3,145,728
#define K_OUT  (N_IN / 4)             // 786,432
#define DEG    16
#define CH     8
#define WAVES  (K_OUT / 16)           // 49,152 waves, one 16-cell tile each

// One wave32 computes a 16x16 D tile per channel with V_WMMA_F32_16X16X4_F32:
//   A[m,d] = M_vals[k0+m, d]          (16x4 per chunk, 4 chunks cover d=0..15)
//   B[d,n] = x[c, M_cols[k0+n, d]]
//   D[m,n] = sum_d A[m,d]*B[d,n]  ->  y[c, k0+m] = D[m,m]
// ISA VGPR striping (32-bit A 16x4 / B 4x16, wave32):
//   lane L: m = n = L&15 ; half-wave hi = L>>4 holds K-pair (hi*2, hi*2+1)
// So each lane needs cols/vals[row, d] for d = 4j + 2*hi + {0,1}  (8B loads).
__global__ __launch_bounds__(256) void sphere_down_wmma(
    const float* __restrict__ x,
    const int*   __restrict__ M_cols,
    const float* __restrict__ M_vals,
    float*       __restrict__ y)
{
    const int lane = threadIdx.x & 31;
    const int wave = (int)((blockIdx.x * blockDim.x + threadIdx.x) >> 5);
    const int k0   = wave << 4;
    const int n    = lane & 15;     // tile row (A's M and B's N for this lane)
    const int hi   = lane >> 4;     // half-wave -> which K pair
    const int row  = k0 + n;

    // ---- stream cols / vals for this lane's (row, K-pairs): 4 x 8B each ----
    int2 cols2[4];
    v2f  avals[4];
#pragma unroll
    for (int j = 0; j < 4; ++j) {
        const int d = j * 4 + hi * 2;
        cols2[j] = *reinterpret_cast<const int2*>(&M_cols[(size_t)row * DEG + d]);
        avals[j] = *reinterpret_cast<const v2f*>(&M_vals[(size_t)row * DEG + d]);
    }

    // Prefetch next tile's index/weight stream (uniform branch; no EXEC change).
    if (wave + 1 < WAVES) {
        __builtin_prefetch(&M_cols[(size_t)(row + 16) * DEG], 0, 3);
        __builtin_prefetch(&M_vals[(size_t)(row + 16) * DEG], 0, 3);
    }

    // Diagonal ownership: D VGPR r holds M=r (lanes 0-15, N=lane) and M=r+8
    // (lanes 16-31, N=lane-16). Diagonal (m,m) lives in lane m (m<8, vgpr m)
    // and lane m+16 (m>=8, vgpr m-8).
    const bool have_diag = (lane < 8) || (lane >= 24);
    const int  vidx = (lane < 8) ? lane : (lane - 24);  // acc element index
    const int  mrow = (lane < 8) ? lane : (lane - 16);  // output m in 0..15

#pragma unroll
    for (int c = 0; c < CH; ++c) {
        const float* __restrict__ xc = x + (size_t)c * N_IN;
        v8f acc = {};
#pragma unroll
        for (int j = 0; j < 4; ++j) {
            v2f b;
            b.x = xc[cols2[j].x];   // gather: clustered around 4*k -> L2 hits
            b.y = xc[cols2[j].y];
            // 8 args: (neg_a, A, neg_b, B, c_mod, C, reuse_a, reuse_b)
            acc = __builtin_amdgcn_wmma_f32_16x16x4_f32(
                false, avals[j], false, b, (short)0, acc, false, false);
        }
        // Select this lane's diagonal element from the 8 acc VGPRs.
        float dval = acc[0];
#pragma unroll
        for (int i = 1; i < 8; ++i)
            if (vidx == i) dval = acc[i];
        // Contiguous 64B store: lanes 0-7 -> k0..k0+7, lanes 24-31 -> k0+8..15.
        if (have_diag)
            y[(size_t)c * K_OUT + k0 + mrow] = dval;
    }
}

extern "C" void kernel_launch(void* const* d_in, const int* in_sizes, int n_in,
                              void* d_out, int out_size, void* d_ws, size_t ws_size,
                              hipStream_t stream) {
    const float* x        = (const float*)d_in[0];
    const int*   M_cols   = (const int*)  d_in[1];
    const float* M_vals   = (const float*)d_in[2];
    const int*   cell_ids = (const int*)  d_in[3];
    float*       y        = (float*)d_out;

    // 49,152 waves / 8 waves-per-block = 6144 blocks, exact (no tail).
    dim3 block(256);
    dim3 grid(WAVES / 8);
    hipLaunchKernelGGL(sphere_down_wmma, grid, block, 0, stream, x, M_cols, M_vals, y);

    // Second tuple output: cell_ids passthrough (bitwise), graph-capture safe.
    hipMemcpyAsync((char*)d_out + (size_t)CH * K_OUT * sizeof(float), cell_ids,
                   (size_t)K_OUT * sizeof(int), hipMemcpyDeviceToDevice, stream);
}